// Encoder_18726057410744
// MI455X (gfx1250) — compile-verified
//
#include <hip/hip_runtime.h>

#define N_NODES 50000
#define D_FEAT 128
#define N_EDGES 800000
#define N_LAYERS 3
#define BN_EPS 1e-5f
#define ROWS_PER_BLOCK 64
#define LDSTRIDE 132   // 528B rows: 16B-aligned for async B128, bank = (4r+c)%64 -> conflict-free

typedef float v2f __attribute__((ext_vector_type(2)));
typedef float v8f __attribute__((ext_vector_type(8)));

// D = A(16x4 f32) * B(4x16 f32) + C(16x16 f32)  -- fp32 WMMA, exact vs reference
__device__ __forceinline__ v8f wmma_f32_16x16x4(v2f a, v2f b, v8f c) {
  return __builtin_amdgcn_wmma_f32_16x16x4_f32(
      /*neg_a=*/false, a, /*neg_b=*/false, b,
      /*c_mod=*/(short)0, c, /*reuse_a=*/false, /*reuse_b=*/false);
}

// Generic LDS pointer -> wave-relative LDS byte offset (low 32 bits of the
// shared aperture address are the LDS offset per the CDNA5 aperture rules).
__device__ __forceinline__ unsigned lds_off32(const void* p) {
  return (unsigned)(unsigned long long)(uintptr_t)p;
}

// CDNA5 async memory->LDS copy (ASYNCcnt-tracked, no VGPR round-trip).
__device__ __forceinline__ void async_ld_b128(unsigned lds_off, const void* g) {
  asm volatile("global_load_async_to_lds_b128 %0, %1, off"
               :: "v"(lds_off), "v"((unsigned long long)(uintptr_t)g)
               : "memory");
}
__device__ __forceinline__ void wait_async0() {
  asm volatile("s_wait_asynccnt 0x0" ::: "memory");
}

// ---------------------------------------------------------------------------
// Edge scatter: agg[dst] += z[src] * w.  One wave per edge; lane q handles
// 4 consecutive features (float4 gather, 4 float atomics).
// ---------------------------------------------------------------------------
__global__ void __launch_bounds__(256)
scatter_kernel(const float* __restrict__ z, const float* __restrict__ ew,
               const int* __restrict__ ei, float* __restrict__ agg) {
  const int t = blockIdx.x * blockDim.x + threadIdx.x;
  const int e = t >> 5;
  const int q = t & 31;
  if (e >= N_EDGES) return;
  const int src = ei[e];
  const int dst = ei[N_EDGES + e];
  const float w = ew[e];
  const float4 v = ((const float4*)(z + (size_t)src * D_FEAT))[q];
  float* ar = agg + (size_t)dst * D_FEAT + q * 4;
  atomicAdd(ar + 0, v.x * w);
  atomicAdd(ar + 1, v.y * w);
  atomicAdd(ar + 2, v.z * w);
  atomicAdd(ar + 3, v.w * w);
}

// ---------------------------------------------------------------------------
// Fused GIN MLP: h=(1+eps)z+agg ; z' = relu(relu(h@W1+b1)@W2+b2)
// Block = 128 threads (4 waves); each wave owns a 16-row strip (full 128 cols).
// z/agg tiles staged with CDNA5 async-to-LDS copies; h combined in place.
// Dynamic LDS: 2 * 64 * 132 * 4 = 67584 B (well within 320KB/WGP).
// ---------------------------------------------------------------------------
__global__ void __launch_bounds__(128)
gin_mlp_kernel(const float* __restrict__ z, const float* __restrict__ agg,
               const float* __restrict__ W1, const float* __restrict__ b1,
               const float* __restrict__ W2, const float* __restrict__ b2,
               const float* __restrict__ eps_arr, int layer,
               float* __restrict__ zout) {
  extern __shared__ float smem[];
  float* sZ = smem;                              // [64][132]
  float* sA = smem + ROWS_PER_BLOCK * LDSTRIDE;  // [64][132]
  const int tid = threadIdx.x;
  const int wv   = tid >> 5;
  const int lane = tid & 31;
  const int hh   = lane >> 4;   // half-wave select (K split / M split)
  const int mm   = lane & 15;
  const int row0 = blockIdx.x * ROWS_PER_BLOCK;

  // Async-stage raw z and agg tiles (each lane moves 16B per copy).
  for (int i = tid; i < ROWS_PER_BLOCK * (D_FEAT / 4); i += 128) {
    const int r = i >> 5, c4 = i & 31;
    int gr = row0 + r;
    if (gr >= N_NODES) gr = N_NODES - 1;
    async_ld_b128(lds_off32(sZ + r * LDSTRIDE + c4 * 4),
                  z + (size_t)gr * D_FEAT + c4 * 4);
    async_ld_b128(lds_off32(sA + r * LDSTRIDE + c4 * 4),
                  agg + (size_t)gr * D_FEAT + c4 * 4);
  }
  // Warm W2 into cache while the async DMA runs (global_prefetch_b8).
#pragma unroll
  for (int j = 0; j < 4; ++j)
    __builtin_prefetch(W2 + ((size_t)tid * 4 + j) * 32, 0, 0);

  const float ce = 1.0f + eps_arr[layer];
  wait_async0();       // this wave's async copies done
  __syncthreads();     // -> everyone's copies visible

  // Combine in place: h = (1+eps)*z + agg
  for (int i = tid; i < ROWS_PER_BLOCK * (D_FEAT / 4); i += 128) {
    const int r = i >> 5, c4 = i & 31;
    float4 zv = *(const float4*)(sZ + r * LDSTRIDE + c4 * 4);
    float4 av = *(const float4*)(sA + r * LDSTRIDE + c4 * 4);
    float4 h;
    h.x = ce * zv.x + av.x;
    h.y = ce * zv.y + av.y;
    h.z = ce * zv.z + av.z;
    h.w = ce * zv.w + av.w;
    *(float4*)(sZ + r * LDSTRIDE + c4 * 4) = h;
  }
  __syncthreads();

  const int wrow = wv * 16;
  const v8f vzero = {0.f, 0.f, 0.f, 0.f, 0.f, 0.f, 0.f, 0.f};
  v8f acc[8];
#pragma unroll
  for (int t = 0; t < 8; ++t) acc[t] = vzero;

  // GEMM1: A from LDS, B = W1 streamed from global (L2-resident)
  for (int k0 = 0; k0 < D_FEAT; k0 += 4) {
    v2f a;
    a.x = sZ[(wrow + mm) * LDSTRIDE + k0 + 2 * hh + 0];
    a.y = sZ[(wrow + mm) * LDSTRIDE + k0 + 2 * hh + 1];
    const float* wk = W1 + (size_t)(k0 + 2 * hh) * D_FEAT + mm;
#pragma unroll
    for (int t = 0; t < 8; ++t) {
      v2f b;
      b.x = wk[t * 16];
      b.y = wk[D_FEAT + t * 16];
      acc[t] = wmma_f32_16x16x4(a, b, acc[t]);
    }
  }

  // bias + ReLU -> H1 written back into this wave's own LDS rows (no barrier needed)
#pragma unroll
  for (int t = 0; t < 8; ++t) {
    const int n = t * 16 + mm;
    const float bias = b1[n];
#pragma unroll
    for (int v = 0; v < 8; ++v) {
      float val = acc[t][v] + bias;
      sZ[(wrow + v + 8 * hh) * LDSTRIDE + n] = val > 0.f ? val : 0.f;
    }
    acc[t] = vzero;
  }

  // GEMM2: A = H1 from LDS, B = W2
  for (int k0 = 0; k0 < D_FEAT; k0 += 4) {
    v2f a;
    a.x = sZ[(wrow + mm) * LDSTRIDE + k0 + 2 * hh + 0];
    a.y = sZ[(wrow + mm) * LDSTRIDE + k0 + 2 * hh + 1];
    const float* wk = W2 + (size_t)(k0 + 2 * hh) * D_FEAT + mm;
#pragma unroll
    for (int t = 0; t < 8; ++t) {
      v2f b;
      b.x = wk[t * 16];
      b.y = wk[D_FEAT + t * 16];
      acc[t] = wmma_f32_16x16x4(a, b, acc[t]);
    }
  }

#pragma unroll
  for (int t = 0; t < 8; ++t) {
    const int n = t * 16 + mm;
    const float bias = b2[n];
#pragma unroll
    for (int v = 0; v < 8; ++v) {
      const int gr = row0 + wrow + v + 8 * hh;
      if (gr < N_NODES) {
        float val = acc[t][v] + bias;
        zout[(size_t)gr * D_FEAT + n] = val > 0.f ? val : 0.f;
      }
    }
  }
}

// ---------------------------------------------------------------------------
// BN statistics: per-column sum / sum-of-squares, coalesced row reads,
// per-block partials merged with float atomics.
// ---------------------------------------------------------------------------
__global__ void __launch_bounds__(128)
bn_stats_kernel(const float* __restrict__ z, float* __restrict__ sums) {
  const int c = threadIdx.x;  // 0..127 = feature column
  float s = 0.f, ss = 0.f;
  for (int r = blockIdx.x; r < N_NODES; r += gridDim.x) {
    const float v = z[(size_t)r * D_FEAT + c];
    s += v;
    ss += v * v;
  }
  atomicAdd(&sums[c], s);
  atomicAdd(&sums[D_FEAT + c], ss);
}

// ---------------------------------------------------------------------------
// Fused BN-normalize (emits zn) + projection GEMM (zn @ Wp + bp) + PReLU.
// z tile async-staged to LDS; per-column scale/shift precomputed in LDS;
// normalization applied in place, then WMMA GEMM.
// Dynamic LDS: 64*132*4 + 2*128*4 = 34816 B.
// ---------------------------------------------------------------------------
__global__ void __launch_bounds__(128)
bn_proj_kernel(const float* __restrict__ z, const float* __restrict__ sums,
               const float* __restrict__ gamma, const float* __restrict__ beta,
               const float* __restrict__ Wp, const float* __restrict__ bp,
               const float* __restrict__ prelu_a,
               float* __restrict__ zn_out, float* __restrict__ p_out) {
  extern __shared__ float smem[];
  float* sZ     = smem;                              // [64][132]
  float* sscale = smem + ROWS_PER_BLOCK * LDSTRIDE;  // [128]
  float* sshift = sscale + D_FEAT;                   // [128]
  const int tid = threadIdx.x;
  const int wv   = tid >> 5;
  const int lane = tid & 31;
  const int hh   = lane >> 4;
  const int mm   = lane & 15;
  const int row0 = blockIdx.x * ROWS_PER_BLOCK;
  const float invN = 1.0f / (float)N_NODES;

  // Async-stage raw z tile.
  for (int i = tid; i < ROWS_PER_BLOCK * (D_FEAT / 4); i += 128) {
    const int r = i >> 5, c4 = i & 31;
    int gr = row0 + r;
    if (gr >= N_NODES) gr = N_NODES - 1;
    async_ld_b128(lds_off32(sZ + r * LDSTRIDE + c4 * 4),
                  z + (size_t)gr * D_FEAT + c4 * 4);
  }
  // Per-column BN scale/shift while DMA runs.
  {
    const float mean = sums[tid] * invN;
    const float var  = sums[D_FEAT + tid] * invN - mean * mean;
    const float rs   = rsqrtf(var + BN_EPS);
    const float sc   = rs * gamma[tid];
    sscale[tid] = sc;
    sshift[tid] = beta[tid] - mean * sc;
  }
  wait_async0();
  __syncthreads();

  // Normalize in place + emit zn.
  for (int i = tid; i < ROWS_PER_BLOCK * (D_FEAT / 4); i += 128) {
    const int r = i >> 5, c4 = i & 31;
    const int gr = row0 + r;
    const int c = c4 * 4;
    float4 zv = *(const float4*)(sZ + r * LDSTRIDE + c);
    float4 o;
    o.x = zv.x * sscale[c + 0] + sshift[c + 0];
    o.y = zv.y * sscale[c + 1] + sshift[c + 1];
    o.z = zv.z * sscale[c + 2] + sshift[c + 2];
    o.w = zv.w * sscale[c + 3] + sshift[c + 3];
    *(float4*)(sZ + r * LDSTRIDE + c) = o;
    if (gr < N_NODES) ((float4*)(zn_out + (size_t)gr * D_FEAT))[c4] = o;
  }
  __syncthreads();

  const int wrow = wv * 16;
  const v8f vzero = {0.f, 0.f, 0.f, 0.f, 0.f, 0.f, 0.f, 0.f};
  v8f acc[8];
#pragma unroll
  for (int t = 0; t < 8; ++t) acc[t] = vzero;

  for (int k0 = 0; k0 < D_FEAT; k0 += 4) {
    v2f a;
    a.x = sZ[(wrow + mm) * LDSTRIDE + k0 + 2 * hh + 0];
    a.y = sZ[(wrow + mm) * LDSTRIDE + k0 + 2 * hh + 1];
    const float* wk = Wp + (size_t)(k0 + 2 * hh) * D_FEAT + mm;
#pragma unroll
    for (int t = 0; t < 8; ++t) {
      v2f b;
      b.x = wk[t * 16];
      b.y = wk[D_FEAT + t * 16];
      acc[t] = wmma_f32_16x16x4(a, b, acc[t]);
    }
  }

  const float slope = prelu_a[0];
#pragma unroll
  for (int t = 0; t < 8; ++t) {
    const int n = t * 16 + mm;
    const float bias = bp[n];
#pragma unroll
    for (int v = 0; v < 8; ++v) {
      const int gr = row0 + wrow + v + 8 * hh;
      if (gr < N_NODES) {
        float val = acc[t][v] + bias;
        p_out[(size_t)gr * D_FEAT + n] = (val >= 0.f) ? val : slope * val;
      }
    }
  }
}

// ---------------------------------------------------------------------------
extern "C" void kernel_launch(void* const* d_in, const int* in_sizes, int n_in,
                              void* d_out, int out_size, void* d_ws, size_t ws_size,
                              hipStream_t stream) {
  const float* x       = (const float*)d_in[0];
  const float* ew      = (const float*)d_in[1];
  const float* W1s     = (const float*)d_in[2];
  const float* b1s     = (const float*)d_in[3];
  const float* W2s     = (const float*)d_in[4];
  const float* b2s     = (const float*)d_in[5];
  const float* eps     = (const float*)d_in[6];
  const float* gamma   = (const float*)d_in[7];
  const float* beta    = (const float*)d_in[8];
  const float* Wp      = (const float*)d_in[9];
  const float* bp      = (const float*)d_in[10];
  const float* prelu_a = (const float*)d_in[11];
  const int*   ei      = (const int*)d_in[12];

  const size_t nd = (size_t)N_NODES * D_FEAT;
  float* zA   = (float*)d_ws;
  float* zB   = zA + nd;
  float* agg  = zB + nd;
  float* sums = agg + nd;

  float* zn_out = (float*)d_out;
  float* p_out  = zn_out + nd;

  const dim3 sblk(256), sgrd((N_EDGES * 32 + 255) / 256);
  const dim3 mblk(128), mgrd((N_NODES + ROWS_PER_BLOCK - 1) / ROWS_PER_BLOCK);
  const size_t mlp_lds = (size_t)2 * ROWS_PER_BLOCK * LDSTRIDE * sizeof(float);
  const size_t bnp_lds = (size_t)ROWS_PER_BLOCK * LDSTRIDE * sizeof(float) +
                         2 * D_FEAT * sizeof(float);

  const float* zin = x;
  float* zout = zA;
  for (int layer = 0; layer < N_LAYERS; ++layer) {
    hipMemsetAsync(agg, 0, nd * sizeof(float), stream);
    scatter_kernel<<<sgrd, sblk, 0, stream>>>(zin, ew, ei, agg);
    gin_mlp_kernel<<<mgrd, mblk, mlp_lds, stream>>>(
        zin, agg,
        W1s + (size_t)layer * D_FEAT * D_FEAT, b1s + (size_t)layer * D_FEAT,
        W2s + (size_t)layer * D_FEAT * D_FEAT, b2s + (size_t)layer * D_FEAT,
        eps, layer, zout);
    zin = zout;
    zout = (zout == zA) ? zB : zA;
  }

  hipMemsetAsync(sums, 0, 2 * D_FEAT * sizeof(float), stream);
  bn_stats_kernel<<<dim3(256), dim3(128), 0, stream>>>(zin, sums);
  bn_proj_kernel<<<mgrd, mblk, bnp_lds, stream>>>(zin, sums, gamma, beta, Wp, bp,
                                                  prelu_a, zn_out, p_out);
}